// Contour_to_distance_map_77549929497023
// MI455X (gfx1250) — compile-verified
//
#include <hip/hip_runtime.h>
#include <math.h>

typedef __attribute__((ext_vector_type(2))) float v2f;
typedef __attribute__((ext_vector_type(8))) float v8f;

#define SIZE_DM 256
#define KVERT 64
#define NSHAPE 8
#define KSIGN 100000.0f
#define EPSV 1e-5f
#define INV_2PI 0.15915494309189535f
#define PI_F 3.14159265358979323846f

__global__ void cdm_init_ws(unsigned* ws) {
    if (threadIdx.x == 0 && blockIdx.x == 0) ws[0] = 0u;
}

// Branch-free acos: Abramowitz-Stegun 4.4.46 (degree-7, |err| <= 2e-8)
__device__ __forceinline__ float acos_fast(float x) {
    float ax = fabsf(x);
    float p = -0.0012624911f;
    p = fmaf(p, ax, 0.0066700901f);
    p = fmaf(p, ax, -0.0170881256f);
    p = fmaf(p, ax, 0.0308918810f);
    p = fmaf(p, ax, -0.0501743046f);
    p = fmaf(p, ax, 0.0889789874f);
    p = fmaf(p, ax, -0.2145988016f);
    p = fmaf(p, ax, 1.5707963050f);
    float r = p * __builtin_amdgcn_sqrtf(1.0f - ax);
    return (x >= 0.0f) ? r : (PI_F - r);
}

// Branch-free tanh: tanh(t) = sign(t) * (1-e)/(1+e), e = exp(-2|t|)
__device__ __forceinline__ float tanh_fast(float t) {
    float e = __expf(-2.0f * fabsf(t));          // v_exp_f32, no range-reduction branches
    float sa = (1.0f - e) * __builtin_amdgcn_rcpf(1.0f + e);
    return copysignf(sa, t);
}

// One wave: 16 consecutive pixels (same image row) x all 64 vertices.
// 4 vertex tiles x 4 WMMA_F32_16X16X4_F32 each (d2, d2_roll, dot, cross).
__global__ __launch_bounds__(256) void cdm_main(const float* __restrict__ contour,
                                                float* __restrict__ out,
                                                unsigned* __restrict__ wsmax) {
    __shared__ float sc[2 * KVERT];   // this block's shape contour (x,y pairs)
    __shared__ float smax[16];

    const int tid       = threadIdx.x;
    const int waveInBlk = tid >> 5;
    const int lane      = tid & 31;
    const bool hi       = (lane & 16) != 0;  // lanes 16-31 hold K=2,3
    const int l15       = lane & 15;

    const int waveId  = blockIdx.x * 8 + waveInBlk;
    const int shape   = waveId >> 12;        // 4096 pixel-groups per shape
    const int group   = waveId & 4095;
    const int basePix = group << 4;          // 16 pixels, all in one image row
    const int row     = basePix >> 8;
    const int col0    = basePix & 255;

    // stage contour for this shape (all 8 waves in the block share one shape)
    if (tid < 2 * KVERT) sc[tid] = contour[shape * (2 * KVERT) + tid];
    __syncthreads();

    const float invS = 1.0f / (float)SIZE_DM;
    // A operand: pixel M = l15, features [mx, my, mx^2+my^2, 1]
    const float mx = (float)row * invS;
    const float my = (float)(col0 + l15) * invS;
    v2f a;
    a.x = hi ? (mx * mx + my * my) : mx;
    a.y = hi ? 1.0f : my;

    float wind[8], mind2[8];
#pragma unroll
    for (int v = 0; v < 8; ++v) { wind[v] = 0.0f; mind2[v] = 3.4e38f; }

#pragma unroll
    for (int t = 0; t < 4; ++t) {
        const int k  = (t << 4) | l15;          // this lane's vertex column
        const int kq = (k + 1) & (KVERT - 1);   // rolled vertex
        const float ckx = sc[2 * k],  cky = sc[2 * k + 1];
        const float cqx = sc[2 * kq], cqy = sc[2 * kq + 1];

        // Branch-free B operands: lanes 0-15 carry K=0,1; lanes 16-31 carry K=2,3.
        v2f b_d2, b_dr, b_dot, b_cr;
        b_d2.x  = hi ? 1.0f                     : -2.0f * ckx;
        b_d2.y  = hi ? (ckx * ckx + cky * cky)  : -2.0f * cky;
        b_dr.x  = hi ? 1.0f                     : -2.0f * cqx;
        b_dr.y  = hi ? (cqx * cqx + cqy * cqy)  : -2.0f * cqy;
        b_dot.x = hi ? 1.0f                     : -(ckx + cqx);
        b_dot.y = hi ? (ckx * cqx + cky * cqy)  : -(cky + cqy);
        b_cr.x  = hi ? 0.0f                     : (cqy - cky);
        b_cr.y  = hi ? (cky * cqx - ckx * cqy)  : (ckx - cqx);

        v8f d2 = {}, dr = {}, dt = {}, cr = {};
        d2 = __builtin_amdgcn_wmma_f32_16x16x4_f32(false, a, false, b_d2,  (short)0, d2, false, false);
        dr = __builtin_amdgcn_wmma_f32_16x16x4_f32(false, a, false, b_dr,  (short)0, dr, false, false);
        dt = __builtin_amdgcn_wmma_f32_16x16x4_f32(false, a, false, b_dot, (short)0, dt, false, false);
        cr = __builtin_amdgcn_wmma_f32_16x16x4_f32(false, a, false, b_cr,  (short)0, cr, false, false);

#pragma unroll
        for (int v = 0; v < 8; ++v) {
            float d2c = fmaxf(d2[v], 0.0f);
            float drc = fmaxf(dr[v], 0.0f);
            mind2[v]  = fminf(mind2[v], d2c);   // sqrt deferred: min commutes with sqrt
            // 1/max(||d||, eps) == min(rsq(d2), 1/eps)   (rsq(0)=inf -> clamped)
            float invnd = fminf(__builtin_amdgcn_rsqf(d2c), 1.0e5f);
            float invnr = fminf(__builtin_amdgcn_rsqf(drc), 1.0e5f);
            float cosv = dt[v] * invnd * invnr;
            cosv = fminf(fmaxf(cosv, -1.0f + EPSV), 1.0f - EPSV);
            float ang = acos_fast(cosv);
            float s   = tanh_fast(KSIGN * cr[v]);
            wind[v]   = fmaf(s, ang, wind[v]);
        }
    }

    // reduce over the 16 vertex columns held across each 16-lane half
#pragma unroll
    for (int m = 1; m < 16; m <<= 1) {
#pragma unroll
        for (int v = 0; v < 8; ++v) {
            wind[v]  += __shfl_xor(wind[v], m, 32);
            mind2[v]  = fminf(mind2[v], __shfl_xor(mind2[v], m, 32));
        }
    }

    float pm = 0.0f;
    if (l15 == 0) {
        // half 0 holds pixel rows 0..7, half 1 rows 8..15 of this group
        const int outBase = shape * (SIZE_DM * SIZE_DM) + basePix + (hi ? 8 : 0);
        float prod[8];
#pragma unroll
        for (int v = 0; v < 8; ++v) {
            float mind = __builtin_amdgcn_sqrtf(mind2[v]);   // single sqrt after reduction
            prod[v] = fabsf(wind[v] * INV_2PI) * mind;
            pm = fmaxf(pm, prod[v]);
        }
        float4* o4 = (float4*)(out + outBase);
        o4[0] = make_float4(prod[0], prod[1], prod[2], prod[3]);
        o4[1] = make_float4(prod[4], prod[5], prod[6], prod[7]);
        smax[waveInBlk * 2 + (hi ? 1 : 0)] = pm;
    }
    __syncthreads();
    if (tid == 0) {
        float bm = smax[0];
#pragma unroll
        for (int i = 1; i < 16; ++i) bm = fmaxf(bm, smax[i]);
        atomicMax(wsmax, __float_as_uint(bm));  // all values >= 0: bit-compare == float-compare
    }
}

__global__ void cdm_norm(float* __restrict__ out, const unsigned* __restrict__ wsmax, int n) {
    int i = blockIdx.x * blockDim.x + threadIdx.x;
    if (i < n) out[i] = out[i] * __builtin_amdgcn_rcpf(__uint_as_float(wsmax[0]));
}

extern "C" void kernel_launch(void* const* d_in, const int* in_sizes, int n_in,
                              void* d_out, int out_size, void* d_ws, size_t ws_size,
                              hipStream_t stream) {
    const float* contour = (const float*)d_in[0];   // (2,4,64,2) fp32
    float* out = (float*)d_out;                     // (2,4,256,256) fp32
    unsigned* ws = (unsigned*)d_ws;

    cdm_init_ws<<<1, 64, 0, stream>>>(ws);

    // 8 shapes * 4096 pixel-groups = 32768 waves; 8 waves per block
    cdm_main<<<4096, 256, 0, stream>>>(contour, out, ws);

    const int n = NSHAPE * SIZE_DM * SIZE_DM;
    cdm_norm<<<(n + 255) / 256, 256, 0, stream>>>(out, ws, n);
}